// DrugModulatedRFALayer_78769700208721
// MI455X (gfx1250) — compile-verified
//
#include <hip/hip_runtime.h>
#include <math.h>

#define N_NODES 4096
#define BATCH   4
#define FDIM    256
#define ODIM    256
#define KNB     15
#define NEG_INF -1000000000.0f
#define ALPHA   0.5f

typedef __attribute__((ext_vector_type(2))) float v2f;
typedef __attribute__((ext_vector_type(8))) float v8f;

// ---------------------------------------------------------------------------
// Kernel 1: importance[b*N+n] = dot(features[b,n,:], attn_kernel[:,0])
// One wave32 per row; coalesced strided loads + shfl reduction.
// ---------------------------------------------------------------------------
__global__ void importance_kernel(const float* __restrict__ features,
                                  const float* __restrict__ attnk,
                                  float* __restrict__ importance) {
  const int wave = threadIdx.x >> 5;
  const int lane = threadIdx.x & 31;
  const int row  = blockIdx.x * 8 + wave;          // [0, B*N)
  const float* f = features + (size_t)row * FDIM;
  float acc = 0.f;
#pragma unroll
  for (int c = 0; c < FDIM; c += 32)
    acc += f[c + lane] * attnk[c + lane];
  for (int off = 16; off >= 1; off >>= 1)
    acc += __shfl_down(acc, off, 32);
  if (lane == 0) importance[row] = acc;
}

// ---------------------------------------------------------------------------
// Kernel 2: support = features @ kernel  via V_WMMA_F32_16X16X4_F32 (exact f32)
// features flattened [B*N, F] = [16384, 256], weight [256, 256].
// One wave per 16x16 C tile; K loop of 64 WMMAs.
// A layout (16x4 f32): lane m = lane&15; VGPR v holds K = kbase + v + 2*(lane>>4)
// B layout (4x16 f32): lane n = lane&15; VGPR v holds K = kbase + v + 2*(lane>>4)
// C layout (16x16 f32): n = lane&15, m = r + 8*(lane>>4) for VGPR r.
// ---------------------------------------------------------------------------
__global__ void support_gemm_kernel(const float* __restrict__ features,
                                    const float* __restrict__ weight,
                                    float* __restrict__ support) {
  const int wave   = threadIdx.x >> 5;
  const int lane   = threadIdx.x & 31;
  const int tile   = blockIdx.x * 8 + wave;        // 16384 tiles
  const int tile_n = tile & 15;                    // 256/16 = 16 n-tiles
  const int tile_m = tile >> 4;                    // 16384/16 = 1024 m-tiles
  const int m0 = tile_m * 16, n0 = tile_n * 16;

  const int half = lane >> 4;                      // 0 or 1
  const int l16  = lane & 15;

  v8f acc = {};
  const float* arow = features + (size_t)(m0 + l16) * FDIM;
#pragma unroll 8
  for (int k = 0; k < FDIM; k += 4) {
    const int ka = k + 2 * half;
    v2f a;
    a.x = arow[ka];
    a.y = arow[ka + 1];
    v2f b;
    b.x = weight[(size_t)ka * ODIM + n0 + l16];
    b.y = weight[(size_t)(ka + 1) * ODIM + n0 + l16];
    acc = __builtin_amdgcn_wmma_f32_16x16x4_f32(
        /*neg_a=*/false, a, /*neg_b=*/false, b,
        /*c_mod=*/(short)0, acc, /*reuse_a=*/false, /*reuse_b=*/false);
  }
#pragma unroll
  for (int r = 0; r < 8; ++r) {
    const int m = r + 8 * half;
    support[(size_t)(m0 + m) * ODIM + n0 + l16] = acc[r];
  }
}

// ---------------------------------------------------------------------------
// Kernel 3: per node i (one 256-thread block), loop over batches:
//   keys[j] = sortable(score(b,i,j)) << 32 | ~j        (lowest-index tiebreak)
//   15 x block-argmax-extract  ->  softmax  ->  gather-aggregate -> ReLU
// ---------------------------------------------------------------------------
__global__ void topk_aggregate_kernel(const float* __restrict__ adj,
                                      const float* __restrict__ features,
                                      const float* __restrict__ importance,
                                      const float* __restrict__ support,
                                      const float* __restrict__ bias,
                                      float* __restrict__ out) {
  __shared__ unsigned long long keys[N_NODES];   // 32 KB
  __shared__ unsigned long long red[256];        // 2 KB
  __shared__ float wts[KNB];
  __shared__ float tval[KNB];
  __shared__ int   tidx[KNB];

  const int i = blockIdx.x;
  const int t = threadIdx.x;
  const float* arow = adj + (size_t)i * N_NODES;

  for (int b = 0; b < BATCH; ++b) {
    const float* impb = importance + b * N_NODES;
    const float imp_i = impb[i];

    // Build sortable keys (value-major, ~index minor for lowest-index ties).
#pragma unroll 4
    for (int c = 0; c < N_NODES; c += 256) {
      const int j = c + t;
      const float score = (arow[j] > 0.f) ? (imp_i + impb[j]) : NEG_INF;
      unsigned int sb = __float_as_uint(score);
      sb = (sb & 0x80000000u) ? ~sb : (sb | 0x80000000u);
      keys[j] = ((unsigned long long)sb << 32) | (unsigned int)(~j);
    }
    __syncthreads();

    // 15 argmax extractions.
    for (int kk = 0; kk < KNB; ++kk) {
      unsigned long long best = 0ull;
#pragma unroll 4
      for (int c = 0; c < N_NODES; c += 256) {
        const unsigned long long v = keys[c + t];
        if (v > best) best = v;
      }
      red[t] = best;
      __syncthreads();
      for (int s = 128; s >= 1; s >>= 1) {
        if (t < s) { if (red[t + s] > red[t]) red[t] = red[t + s]; }
        __syncthreads();
      }
      if (t == 0) {
        const unsigned long long w = red[0];
        const int j = (int)(~(unsigned int)w);
        unsigned int sb = (unsigned int)(w >> 32);
        sb = (sb & 0x80000000u) ? (sb & 0x7FFFFFFFu) : ~sb;
        tval[kk] = __uint_as_float(sb);
        tidx[kk] = j;
        keys[j]  = 0ull;                 // exclude from next extraction
      }
      __syncthreads();
    }

    // Softmax over the 15 top values (shift-invariant; matches reference).
    if (t == 0) {
      float m = tval[0];
      for (int kk = 1; kk < KNB; ++kk) m = fmaxf(m, tval[kk]);
      float s = 0.f, e[KNB];
      for (int kk = 0; kk < KNB; ++kk) { e[kk] = __expf(tval[kk] - m); s += e[kk]; }
      const float inv = 1.f / s;
      for (int kk = 0; kk < KNB; ++kk) wts[kk] = e[kk] * inv;
    }
    __syncthreads();

    // Aggregate: thread t == feature f (ODIM == blockDim.x == 256).
    {
      const int f = t;
      float acc = 0.f;
#pragma unroll
      for (int kk = 0; kk < KNB; ++kk)
        acc += wts[kk] * support[((size_t)b * N_NODES + tidx[kk]) * ODIM + f];
      const float o = ALPHA * features[((size_t)b * N_NODES + i) * FDIM + f]
                    + (1.f - ALPHA) * acc + bias[f];
      out[((size_t)b * N_NODES + i) * ODIM + f] = fmaxf(o, 0.f);
    }
    __syncthreads();
  }
}

// ---------------------------------------------------------------------------
extern "C" void kernel_launch(void* const* d_in, const int* in_sizes, int n_in,
                              void* d_out, int out_size, void* d_ws, size_t ws_size,
                              hipStream_t stream) {
  const float* adj      = (const float*)d_in[0];   // [4096,4096]
  const float* features = (const float*)d_in[1];   // [4,4096,256]
  const float* attnk    = (const float*)d_in[2];   // [256,1]
  const float* weight   = (const float*)d_in[3];   // [256,256]
  const float* bias     = (const float*)d_in[4];   // [256]
  float* out = (float*)d_out;

  float* support    = (float*)d_ws;                               // 16384*256 f32
  float* importance = support + (size_t)BATCH * N_NODES * ODIM;   // 16384 f32

  const int rows = BATCH * N_NODES;                // 16384
  importance_kernel<<<rows / 8, 256, 0, stream>>>(features, attnk, importance);

  const int tiles = (rows / 16) * (ODIM / 16);     // 16384 tiles
  support_gemm_kernel<<<tiles / 8, 256, 0, stream>>>(features, weight, support);

  topk_aggregate_kernel<<<N_NODES, 256, 0, stream>>>(adj, features, importance,
                                                     support, bias, out);
}